// Net_15487652069990
// MI455X (gfx1250) — compile-verified
//
#include <hip/hip_runtime.h>

// ---------------- model constants ----------------
#define N_OSC   50
#define N_PERC  40
#define STEPS   700
#define DT      0.001f
#define R_MOM   0.02f
#define L0C     0.1f
#define INERTIA 0.001f
#define DAMP    0.01f
#define K_MUS   50.0f
#define L_REST  0.05f
#define TAU_M   0.05f
#define HALF_PI 1.5707963267948966f

#define KTOT    2000          // N_OSC * N_PERC
#define FC4PACK 64000         // 500 K-chunks * 2 N-tiles * 32 lanes * 2 vals
#define TABS    10000         // enc0, enc1, bias, dec0, dec1 (each KTOT)
#define PACKTOT (FC4PACK + TABS)

typedef float v2f __attribute__((ext_vector_type(2)));
typedef float v8f __attribute__((ext_vector_type(8)));

// ---------------------------------------------------------------------------
// Preamble: per-batch tiny MLP -> osc init states (B x 100) and direct (B x 2)
// One block per batch element, 128 threads.
// ---------------------------------------------------------------------------
__global__ void preamble_kernel(const float* __restrict__ x,
                                const float* __restrict__ fc1w, const float* __restrict__ fc1b,
                                const float* __restrict__ fc2w, const float* __restrict__ fc2b,
                                const float* __restrict__ fc3w, const float* __restrict__ fc3b,
                                const float* __restrict__ fcdw, const float* __restrict__ fcdb,
                                float* __restrict__ osInit, float* __restrict__ direct)
{
    __shared__ float h[128];
    __shared__ float h2[128];
    const int b = blockIdx.x;
    const int j = threadIdx.x;
    const float x0 = x[b * 2 + 0];
    const float x1 = x[b * 2 + 1];

    float hj = fmaf(x1, fc1w[j * 2 + 1], fmaf(x0, fc1w[j * 2 + 0], fc1b[j]));
    h[j] = fmaxf(hj, 0.0f);
    __syncthreads();

    float s = fc2b[j];
    #pragma unroll 8
    for (int i = 0; i < 128; ++i) s = fmaf(fc2w[j * 128 + i], h[i], s);
    h2[j] = fmaxf(s, 0.0f);
    __syncthreads();

    if (j < 100) {
        float t = fc3b[j];
        #pragma unroll 8
        for (int i = 0; i < 128; ++i) t = fmaf(fc3w[j * 128 + i], h2[i], t);
        osInit[b * 100 + j] = t;               // o_state[o][d] = xo[b][2o+d]
    }
    if (j < 2) {
        float t = fcdb[j];
        #pragma unroll 8
        for (int i = 0; i < 128; ++i) t = fmaf(fcdw[j * 128 + i], h[i], t);
        direct[b * 2 + j] = t;
    }
}

// ---------------------------------------------------------------------------
// Pack: swizzle fc4_w into WMMA B-fragment order + flatten enc/bias/dec tables.
//   B frag (f32 16x16x4): VGPR v, lane l -> row K = 4c + v + (l<16?0:2),
//                         col N = 16t + (l&15).  Packed so each lane reads a
//                         contiguous float2 -> conflict-free ds_load_b64.
// ---------------------------------------------------------------------------
__global__ void pack_kernel(const float* __restrict__ enc,
                            const float* __restrict__ oscBias,
                            const float* __restrict__ dec,
                            const float* __restrict__ fc4w,
                            float* __restrict__ wsPack)
{
    const int i = blockIdx.x * blockDim.x + threadIdx.x;
    if (i < FC4PACK) {
        const int c    = i >> 7;          // K-chunk (500)
        const int rem  = i & 127;
        const int t    = rem >> 6;        // N tile (0/1)
        const int r2   = rem & 63;
        const int lane = r2 >> 1;
        const int v    = r2 & 1;
        const int n = t * 16 + (lane & 15);
        const int k = c * 4 + v + ((lane < 16) ? 0 : 2);
        wsPack[i] = fc4w[n * KTOT + k];
    } else if (i < PACKTOT) {
        const int j  = i - FC4PACK;
        const int sel = j / KTOT;
        const int kk  = j % KTOT;
        const int o = kk / N_PERC, p = kk % N_PERC;
        float val;
        if      (sel == 0) val = enc[(o * N_PERC + p) * 2 + 0];
        else if (sel == 1) val = enc[(o * N_PERC + p) * 2 + 1];
        else if (sel == 2) val = oscBias[o * N_PERC + p];
        else if (sel == 3) val = dec[(o * 2 + 0) * N_PERC + p];
        else               val = dec[(o * 2 + 1) * N_PERC + p];
        wsPack[FC4PACK + j] = val;
    }
}

// ---------------------------------------------------------------------------
// Main 700-step simulation. 64 batch rows per block (4 waves x 16 rows).
// LDS: fc4 B-frags (250KiB) + tables (39KiB) + per-wave osc state (25KiB).
// Per step/wave: 1000x v_wmma_f32_16x16x4_f32 (K=2000, N=32) fused with
// oscillator dynamics; epilogue = fc5 projection via half-wave butterflies
// + Euler limb integration; outputs streamed to global.
// ---------------------------------------------------------------------------
#define SM_TABS 64000
#define SM_OS   74000
#define SM_TOT  (74000 + 4 * 16 * 100)     // 80400 floats = 321600 bytes

__global__ void __launch_bounds__(128)
sim_kernel(const float* __restrict__ x,
           const float* __restrict__ wsPack,
           const float* __restrict__ osInit,
           const float* __restrict__ direct,
           const float* __restrict__ fc4b,
           const float* __restrict__ fc5w,
           const float* __restrict__ fc5b,
           float* __restrict__ outL,
           float* __restrict__ outA,
           int B)
{
    extern __shared__ float smem[];
    const int tid  = threadIdx.x;
    const int lane = tid & 31;
    const int wave = tid >> 5;
    const int b16  = lane & 15;
    const int b0   = blockIdx.x * 64;
    const int b    = b0 + wave * 16 + b16;

    // ---- stage packed weights + tables (linear, coalesced) ----
    for (int i = tid; i < PACKTOT; i += 128) smem[i] = wsPack[i];
    // ---- stage oscillator states for this block's 64 rows ----
    for (int i = tid; i < 64 * 100; i += 128) {
        const int bl = i / 100, j = i % 100;
        smem[SM_OS + (bl >> 4) * 1600 + (bl & 15) * 100 + j] = osInit[(b0 + bl) * 100 + j];
    }
    __syncthreads();

    float* osW = smem + SM_OS + wave * 1600;     // [16 rows][50 osc][2]
    const float* tE0 = smem + SM_TABS;
    const float* tE1 = smem + SM_TABS + KTOT;
    const float* tBi = smem + SM_TABS + 2 * KTOT;
    const float* tD0 = smem + SM_TABS + 3 * KTOT;
    const float* tD1 = smem + SM_TABS + 4 * KTOT;

    // per-lane constants (lane -> output column n_lo = b16, n_hi = 16+b16)
    const float fb0 = fc4b[b16];
    const float fb1 = fc4b[16 + b16];
    const float w00 = fc5w[b16],      w01 = fc5w[16 + b16];
    const float w10 = fc5w[32 + b16], w11 = fc5w[48 + b16];
    const float f5b0 = fc5b[0], f5b1 = fc5b[1];
    const float dir0 = direct[b * 2 + 0];
    const float dir1 = direct[b * 2 + 1];

    // limb state (replicated across wave halves; lanes<16 own the stores)
    float th = x[b * 2 + 0];
    float om = 0.0f;
    float m1 = L0C - R_MOM * th;
    float m2 = L0C + R_MOM * th;

    const int off = (lane < 16) ? 0 : 2;   // A/B fragment K offset per half

    for (int step = 0; step < STEPS; ++step) {
        v8f acc0 = {0.f,0.f,0.f,0.f,0.f,0.f,0.f,0.f};   // b1 cols 0..15
        v8f acc1 = {0.f,0.f,0.f,0.f,0.f,0.f,0.f,0.f};   // b1 cols 16..31

        for (int o = 0; o < N_OSC; ++o) {
            const v2f os = *(const v2f*)(osW + b16 * 100 + o * 2);
            float d0 = 0.0f, d1 = 0.0f;
            #pragma unroll
            for (int c2 = 0; c2 < 10; ++c2) {
                const int k = o * N_PERC + c2 * 4 + off;
                const v2f e0 = *(const v2f*)(tE0 + k);
                const v2f e1 = *(const v2f*)(tE1 + k);
                const v2f bi = *(const v2f*)(tBi + k);
                const v2f q0 = *(const v2f*)(tD0 + k);
                const v2f q1 = *(const v2f*)(tD1 + k);
                // a = relu(enc0*os0 + enc1*os1 + bias), directly in A-frag layout
                const float a0 = fmaxf(fmaf(e1.x, os.y, fmaf(e0.x, os.x, bi.x)), 0.0f);
                const float a1 = fmaxf(fmaf(e1.y, os.y, fmaf(e0.y, os.x, bi.y)), 0.0f);
                v2f A; A.x = a0; A.y = a1;
                const int cg = (o * 10 + c2) * 2;
                const v2f B0 = *(const v2f*)(smem + (cg + 0) * 64 + lane * 2);
                const v2f B1 = *(const v2f*)(smem + (cg + 1) * 64 + lane * 2);
                acc0 = __builtin_amdgcn_wmma_f32_16x16x4_f32(false, A, false, B0,
                                                             (short)0, acc0, false, false);
                acc1 = __builtin_amdgcn_wmma_f32_16x16x4_f32(false, A, false, B1,
                                                             (short)0, acc1, false, false);
                // deriv accumulation reuses the A fragment
                d0 = fmaf(q0.x, a0, fmaf(q0.y, a1, d0));
                d1 = fmaf(q1.x, a0, fmaf(q1.y, a1, d1));
            }
            // combine the two K-halves, then Euler-update this oscillator
            d0 += __shfl_xor(d0, 16);
            d1 += __shfl_xor(d1, 16);
            if (lane < 16) {
                osW[b16 * 100 + o * 2 + 0] = os.x + DT * d0;
                osW[b16 * 100 + o * 2 + 1] = os.y + DT * d1;
            }
        }

        // ---- epilogue: b1 = relu(acc + fc4_b); activation = b1 @ fc5^T + fc5_b + direct
        float act0 = 0.0f, act1 = 0.0f;
        #pragma unroll
        for (int r = 0; r < 8; ++r) {
            const float u0 = fmaxf(acc0[r] + fb0, 0.0f);   // (M=r | r+8, N=lane-half)
            const float u1 = fmaxf(acc1[r] + fb1, 0.0f);
            float p0 = fmaf(u0, w00, u1 * w01);
            float p1 = fmaf(u0, w10, u1 * w11);
            #pragma unroll
            for (int m = 1; m < 16; m <<= 1) {              // reduce within each half
                p0 += __shfl_xor(p0, m);
                p1 += __shfl_xor(p1, m);
            }
            const float lo0 = __shfl(p0, 0),  hi0 = __shfl(p0, 16);
            const float lo1 = __shfl(p1, 0),  hi1 = __shfl(p1, 16);
            if (b16 == r)     { act0 = lo0; act1 = lo1; }   // batch row r
            if (b16 == r + 8) { act0 = hi0; act1 = hi1; }   // batch row r+8
        }
        act0 += f5b0 + dir0;
        act1 += f5b1 + dir1;

        // ---- limb dynamics (explicit Euler + clamp) ----
        const float f1 = fmaxf(act0, 0.0f) * K_MUS * fmaxf(m1 - L_REST, 0.0f);
        const float f2 = fmaxf(act1, 0.0f) * K_MUS * fmaxf(m2 - L_REST, 0.0f);
        const float domega = (R_MOM * (f2 - f1) - DAMP * om) / INERTIA;
        const float dm1 = ((L0C - R_MOM * th) - m1) / TAU_M;
        const float dm2 = ((L0C + R_MOM * th) - m2) / TAU_M;
        const float pos = th + DT * om;
        const float vel = om + DT * domega;
        m1 = m1 + DT * dm1;
        m2 = m2 + DT * dm2;
        const bool inb = (pos > -HALF_PI) && (pos < HALF_PI);
        th = fminf(fmaxf(pos, -HALF_PI), HALF_PI);
        om = inb ? vel : 0.0f;

        if (lane < 16) {
            *(float4*)(outL + ((size_t)step * B + b) * 4) = make_float4(th, om, m1, m2);
            *(float2*)(outA + ((size_t)step * B + b) * 2) = make_float2(act0, act1);
        }
    }
}

// ---------------------------------------------------------------------------
extern "C" void kernel_launch(void* const* d_in, const int* in_sizes, int n_in,
                              void* d_out, int out_size, void* d_ws, size_t ws_size,
                              hipStream_t stream)
{
    const float* x       = (const float*)d_in[0];
    const float* fc1w    = (const float*)d_in[1];
    const float* fc1b    = (const float*)d_in[2];
    const float* fc2w    = (const float*)d_in[3];
    const float* fc2b    = (const float*)d_in[4];
    const float* fc3w    = (const float*)d_in[5];
    const float* fc3b    = (const float*)d_in[6];
    const float* fcdw    = (const float*)d_in[7];
    const float* fcdb    = (const float*)d_in[8];
    const float* enc     = (const float*)d_in[9];
    const float* oscBias = (const float*)d_in[10];
    const float* dec     = (const float*)d_in[11];
    const float* fc4w    = (const float*)d_in[12];
    const float* fc4b    = (const float*)d_in[13];
    const float* fc5w    = (const float*)d_in[14];
    const float* fc5b    = (const float*)d_in[15];

    const int B = in_sizes[0] / 2;           // 4096

    float* ws     = (float*)d_ws;
    float* wsPack = ws;                                   // PACKTOT floats
    float* wsOs   = ws + PACKTOT;                         // B*100
    float* wsDir  = ws + PACKTOT + (size_t)B * 100;       // B*2

    float* outL = (float*)d_out;
    float* outA = outL + (size_t)STEPS * B * 4;

    preamble_kernel<<<B, 128, 0, stream>>>(x, fc1w, fc1b, fc2w, fc2b,
                                           fc3w, fc3b, fcdw, fcdb, wsOs, wsDir);
    pack_kernel<<<(PACKTOT + 255) / 256, 256, 0, stream>>>(enc, oscBias, dec, fc4w, wsPack);

    const size_t smBytes = SM_TOT * sizeof(float);        // 321600 B <= 320 KiB
    (void)hipFuncSetAttribute((const void*)sim_kernel,
                              hipFuncAttributeMaxDynamicSharedMemorySize,
                              (int)smBytes);
    sim_kernel<<<B / 64, 128, smBytes, stream>>>(x, wsPack, wsOs, wsDir,
                                                 fc4b, fc5w, fc5b, outL, outA, B);
}